// SoftCrossEntropyLoss_70635032150275
// MI455X (gfx1250) — compile-verified
//
#include <hip/hip_runtime.h>
#include <stdint.h>

typedef __attribute__((ext_vector_type(2))) float v2f;
typedef __attribute__((ext_vector_type(8))) float v8f;

#define KDIM 512
#define ROWSTR 516                // padded LDS row stride in floats (bank-conflict-free)
#define WBUF (16 * ROWSTR)        // one 16x512 tile (padded), per wave
#define WSTR (2 * WBUF + 16)      // x tile + t tile + rowmax[16], per wave

// one 16-byte async DMA: global -> LDS (CDNA5 GLOBAL_LOAD_ASYNC_TO_LDS_B128)
__device__ __forceinline__ void async_cp16(uint32_t ldsByteAddr, const float* g) {
  asm volatile("global_load_async_to_lds_b128 %0, %1, off"
               :: "v"(ldsByteAddr), "v"(g) : "memory");
}

__global__ void sce_main(const float* __restrict__ x,
                         const float* __restrict__ t,
                         float* __restrict__ partial) {
  extern __shared__ float smem[];
  const int tid  = threadIdx.x;
  const int lane = tid & 31;
  const int wid  = tid >> 5;

  float* xb   = smem + wid * WSTR;   // x tile, padded rows
  float* tb   = xb + WBUF;           // t tile, padded rows
  float* rm   = tb + WBUF;           // rowmax[16]
  float* wsum = smem + 4 * WSTR;     // per-wave tile losses

  const long long tile = (long long)blockIdx.x * 4 + wid;   // 16-row tile index
  const long long base = tile * (16LL * KDIM);

  // ---- stage tiles (each a contiguous 32KB block) into padded LDS via async DMA.
  //      All x transfers are issued before all t transfers; async loads complete
  //      in order, so waiting asynccnt<=63 (after 128 issues) guarantees the x
  //      tile is resident while the t tile is still streaming.
  {
    const float* xg = x + base;
    const float* tg = t + base;
    #pragma unroll 8
    for (int i = 0; i < 64; ++i) {
      int e = i * 128 + lane * 4;          // float4-granular flat element index
      int r = e >> 9;                      // row within tile
      int c = e & (KDIM - 1);              // col within row
      async_cp16((uint32_t)(uintptr_t)(xb + r * ROWSTR + c), xg + e);
    }
    #pragma unroll 8
    for (int i = 0; i < 64; ++i) {
      int e = i * 128 + lane * 4;
      int r = e >> 9;
      int c = e & (KDIM - 1);
      async_cp16((uint32_t)(uintptr_t)(tb + r * ROWSTR + c), tg + e);
    }
    asm volatile("s_wait_asynccnt 0x3f" ::: "memory");   // x tile complete
  }

  // fragment addressing: lane l and lane l+16 jointly cover row (l&15);
  // low half reads cols = {0,1} mod 4, high half cols = {2,3} mod 4
  const int   hi    = lane >> 4;
  const int   myrow = lane & 15;
  const float* xr = xb + myrow * ROWSTR + 2 * hi;
  const float* tr = tb + myrow * ROWSTR + 2 * hi;

  // ---- pass A: per-row max, overlapped with the t-tile DMA tail
  float mrow = -3.402823466e38f;
  #pragma unroll 8
  for (int k0 = 0; k0 < KDIM; k0 += 4) {
    v2f xv = *(const v2f*)(xr + k0);
    mrow = fmaxf(mrow, fmaxf(xv.x, xv.y));
  }
  mrow = fmaxf(mrow, __shfl_xor(mrow, 16));   // merge the two column halves
  if (lane < 16) rm[lane] = mrow;             // needed later for D extraction

  asm volatile("s_wait_asynccnt 0x0" ::: "memory");      // t tile complete

  // ---- pass B: WMMA row-sum accumulation (B = all-ones => D[i][j] = sum_k A[i,k])
  const v2f ones = (v2f){1.0f, 1.0f};
  v8f cSE = {};   // sum exp(x - m)
  v8f cDOT = {};  // sum t*x
  v8f cTS = {};   // sum t
  #pragma unroll 4
  for (int k0 = 0; k0 < KDIM; k0 += 4) {
    v2f xv = *(const v2f*)(xr + k0);
    v2f tv = *(const v2f*)(tr + k0);
    v2f ev;
    ev.x = __expf(xv.x - mrow);
    ev.y = __expf(xv.y - mrow);
    v2f pv = tv * xv;
    cSE  = __builtin_amdgcn_wmma_f32_16x16x4_f32(false, ev, false, ones, (short)0, cSE,  false, false);
    cDOT = __builtin_amdgcn_wmma_f32_16x16x4_f32(false, pv, false, ones, (short)0, cDOT, false, false);
    cTS  = __builtin_amdgcn_wmma_f32_16x16x4_f32(false, tv, false, ones, (short)0, cTS,  false, false);
  }

  // ---- combine: lane l VGPR r holds row (r + 8*hi) sums (replicated across N)
  float s = 0.0f;
  #pragma unroll
  for (int r = 0; r < 8; ++r) {
    float se = cSE[r];
    float dt = cDOT[r];
    float ts = cTS[r];
    float mm = rm[r + 8 * hi];
    s += ts * (mm + __logf(se)) - dt;   // = +loss for this row
  }
  float tileLoss = s + __shfl_xor(s, 16);   // low-half rows + high-half rows
  if (lane == 0) wsum[wid] = tileLoss;
  __syncthreads();
  if (tid == 0) partial[blockIdx.x] = wsum[0] + wsum[1] + wsum[2] + wsum[3];
}

// deterministic fixed-order final reduction of per-block partials
__global__ void sce_reduce(const float* __restrict__ partial, int nPart,
                           float invN, float* __restrict__ out) {
  __shared__ float sm[256];
  int tid = threadIdx.x;
  float s = 0.0f;
  for (int i = tid; i < nPart; i += 256) s += partial[i];
  sm[tid] = s;
  __syncthreads();
  for (int off = 128; off > 0; off >>= 1) {
    if (tid < off) sm[tid] += sm[tid + off];
    __syncthreads();
  }
  if (tid == 0) out[0] = sm[0] * invN;
}

extern "C" void kernel_launch(void* const* d_in, const int* in_sizes, int n_in,
                              void* d_out, int out_size, void* d_ws, size_t ws_size,
                              hipStream_t stream) {
  const float* x = (const float*)d_in[0];
  const float* t = (const float*)d_in[1];
  float* out     = (float*)d_out;
  float* partial = (float*)d_ws;

  const long long total = (long long)in_sizes[0];
  const int n      = (int)(total / KDIM);   // 131072 rows
  const int blocks = n / 64;                // 4 waves/block * 16 rows/wave

  size_t shmem = (size_t)(4 * WSTR + 4) * sizeof(float);  // ~264 KB / WG
  sce_main<<<blocks, 128, shmem, stream>>>(x, t, partial);
  sce_reduce<<<1, 256, 0, stream>>>(partial, blocks, 1.0f / (float)n, out);
}